// PatchPooling_65146063946328
// MI455X (gfx1250) — compile-verified
//
#include <hip/hip_runtime.h>
#include <stdint.h>

// Segment-wise max pooling: out[s][f] = max over rows r with ids[r]==s of patches[r][f]
// patches: 1048576 x 128 f32, ids in [0,256), out: 256 x 128 f32.

#define NSEG      256
#define NFEAT     128
#define ACC_ELEMS (NSEG * NFEAT)      // 32768 u32 accumulator entries
#define ACC_BYTES (ACC_ELEMS * 4)     // 128 KB LDS (fits CDNA5's 320KB WGP LDS)
#define P1_BLOCKS 256
#define P1_THREADS 256                // 8 wave32s per block
#define PF_ROWS   16                  // prefetch distance: 16 rows = 8 KB ahead
#define MONO_NEGINF 0x007FFFFFu       // monotonic encoding of -inf

// Order-preserving f32 -> u32 map: larger float <=> larger unsigned int.
__device__ __forceinline__ uint32_t f2mono(float f) {
  uint32_t b = __float_as_uint(f);
  return b ^ (uint32_t)(((int32_t)b >> 31) | 0x80000000);
}
__device__ __forceinline__ float mono2f(uint32_t m) {
  uint32_t mask = (uint32_t)((int32_t)m >> 31);     // all-ones if top bit set
  return __uint_as_float(m ^ (~mask | 0x80000000u));
}

// Fold one row into the LDS accumulator.
// Lane l handles features {l, 32+l, 64+l, 96+l}: each ds_max_u32 touches 32
// consecutive LDS dwords -> conflict-free across wave32 on 64 banks.
__device__ __forceinline__ void fold_row(uint32_t* __restrict__ acc,
                                         const float* __restrict__ row,
                                         int id, int lane)
{
  const uint32_t base = (uint32_t)id * NFEAT;
#pragma unroll
  for (int k = 0; k < 4; ++k) {
    const uint32_t m = f2mono(row[k * 32 + lane]);          // coalesced 128B load
    atomicMax(&acc[base + (uint32_t)(k * 32 + lane)], m);   // ds_max_u32
  }
}

// Pass 1: each workgroup privatizes the full 256x128 output in 128KB LDS,
// streams a contiguous chunk of rows, folds with ds_max_u32 atomics.
template <bool USE_WS>
__global__ __launch_bounds__(P1_THREADS)
void seg_max_pass1(const float* __restrict__ patches,
                   const int* __restrict__ patch_ids,
                   uint32_t* __restrict__ dst,      // USE_WS: partials; else u-domain d_out
                   int nrows)
{
  extern __shared__ uint32_t acc[];                 // ACC_ELEMS u32
  const int tid  = threadIdx.x;
  const int lane = tid & 31;
  const int wav  = tid >> 5;

  for (int i = tid; i < ACC_ELEMS; i += P1_THREADS) acc[i] = MONO_NEGINF;
  __syncthreads();

  const int waves_total = gridDim.x * (P1_THREADS / 32);
  const int gwave = blockIdx.x * (P1_THREADS / 32) + wav;
  const int chunk = (nrows + waves_total - 1) / waves_total;
  const int r0 = min(nrows, gwave * chunk);
  const int r1 = min(nrows, r0 + chunk);

  // Wave-uniform running state: scalar row index, incrementing row pointer.
  int r = __builtin_amdgcn_readfirstlane(r0);
  const float* row = patches + (size_t)r * NFEAT;
  const int rmain = (r1 - r0 > PF_ROWS) ? (r1 - PF_ROWS) : r0;

  // Main loop: prefetch PF_ROWS ahead, 2 rows per iteration (8 loads in flight).
#pragma unroll 2
  for (; r < rmain; ++r, row += NFEAT) {
    // gfx1250 streaming prefetch (global_prefetch_b8): 32 lanes * 16B cover
    // the full 512B row at distance PF_ROWS.
    __builtin_prefetch((const char*)(row + PF_ROWS * NFEAT) + lane * 16, 0, 1);
    const int id = patch_ids[r];                    // scalar (uniform) load
    fold_row(acc, row, id, lane);
  }
  // Tail: last PF_ROWS rows, no prefetch (stays in-bounds).
  for (; r < r1; ++r, row += NFEAT) {
    const int id = patch_ids[r];
    fold_row(acc, row, id, lane);
  }
  __syncthreads();

  if (USE_WS) {
    uint32_t* out = dst + (size_t)blockIdx.x * ACC_ELEMS;
    for (int i = tid; i < ACC_ELEMS; i += P1_THREADS) out[i] = acc[i];
  } else {
    for (int i = tid; i < ACC_ELEMS; i += P1_THREADS) atomicMax(&dst[i], acc[i]);
  }
}

// Pass 2 (ws path): serially reduce per-block partials with plain loads, decode to f32.
__global__ __launch_bounds__(256)
void seg_max_reduce_ws(const uint32_t* __restrict__ ws, float* __restrict__ out, int nparts)
{
  const int i = blockIdx.x * blockDim.x + threadIdx.x;   // exactly ACC_ELEMS threads
  uint32_t m = MONO_NEGINF;
  for (int b = 0; b < nparts; ++b) {
    const uint32_t v = ws[(size_t)b * ACC_ELEMS + i];    // coalesced 128B/wave reads
    m = v > m ? v : m;
  }
  out[i] = mono2f(m);
}

// Atomic-fallback path helpers: init d_out in u-domain, decode in place at the end.
__global__ __launch_bounds__(256)
void seg_max_init_u(uint32_t* __restrict__ out)
{
  const int i = blockIdx.x * blockDim.x + threadIdx.x;
  if (i < ACC_ELEMS) out[i] = MONO_NEGINF;
}

__global__ __launch_bounds__(256)
void seg_max_finalize_u(float* __restrict__ out)
{
  const int i = blockIdx.x * blockDim.x + threadIdx.x;
  if (i < ACC_ELEMS) {
    const uint32_t m = __float_as_uint(out[i]);  // bits are u-domain here
    out[i] = mono2f(m);
  }
}

extern "C" void kernel_launch(void* const* d_in, const int* in_sizes, int n_in,
                              void* d_out, int out_size, void* d_ws, size_t ws_size,
                              hipStream_t stream)
{
  // setup_inputs order: [0] batch_size (scalar, fixed 256), [1] patches f32, [2] patch_ids i32
  const float* patches   = (const float*)d_in[1];
  const int*   patch_ids = (const int*)d_in[2];
  const int    nrows     = in_sizes[2];            // 1048576
  float*       out       = (float*)d_out;          // 256*128 f32

  const size_t ws_need = (size_t)P1_BLOCKS * ACC_BYTES;   // 32 MB of partials

  if (ws_size >= ws_need) {
    uint32_t* ws = (uint32_t*)d_ws;
    seg_max_pass1<true><<<P1_BLOCKS, P1_THREADS, ACC_BYTES, stream>>>(
        patches, patch_ids, ws, nrows);
    seg_max_reduce_ws<<<ACC_ELEMS / 256, 256, 0, stream>>>(ws, out, P1_BLOCKS);
  } else {
    uint32_t* uo = (uint32_t*)d_out;
    seg_max_init_u<<<ACC_ELEMS / 256, 256, 0, stream>>>(uo);
    seg_max_pass1<false><<<P1_BLOCKS, P1_THREADS, ACC_BYTES, stream>>>(
        patches, patch_ids, uo, nrows);
    seg_max_finalize_u<<<ACC_ELEMS / 256, 256, 0, stream>>>(out);
  }
}